// GATLayer2_13219909337126
// MI455X (gfx1250) — compile-verified
//
#include <hip/hip_runtime.h>
#include <math.h>

// ---------------------------------------------------------------------------
// GAT layer for MI455X (gfx1250): 5 fp32 GEMMs on v_wmma_f32_16x16x4_f32.
// dims: N=8192 queries, M=4096 keys/in_features, D=128 out_features
// ---------------------------------------------------------------------------

typedef float v2f __attribute__((ext_vector_type(2)));
typedef float v8f __attribute__((ext_vector_type(8)));

static constexpr int N = 8192;
static constexpr int M = 4096;
static constexpr int D = 128;

#define LRELU_ALPHA 0.2f
#define KEEP        0.9f
#define INV_KEEP    (1.0f / 0.9f)
#define NEGV        -9e15f
// floor(0.9 * 2^32)
#define KEEP_THRESH 3865470566u
#define SEED1 0x9E3779B9u
#define SEED2 0x85EBCA6Bu

__device__ __forceinline__ v8f wmma_f32(v2f a, v2f b, v8f c) {
    // D = A(16x4,f32) * B(4x16,f32) + C(16x16,f32)
    return __builtin_amdgcn_wmma_f32_16x16x4_f32(
        /*neg_a=*/false, a, /*neg_b=*/false, b,
        /*c_mod=*/(short)0, c, /*reuse_a=*/false, /*reuse_b=*/false);
}

// deterministic stateless dropout multiplier (lowbias32 hash)
__device__ __forceinline__ float dropmul(unsigned idx, unsigned seed) {
    unsigned x = idx ^ seed;
    x ^= x >> 16; x *= 0x7feb352du;
    x ^= x >> 15; x *= 0x846ca68bu;
    x ^= x >> 16;
    return (x < KEEP_THRESH) ? INV_KEEP : 0.0f;
}

// ---------------------------------------------------------------------------
// Kernel 1: Wh[N,D] = h_s[N,M] @ W[M,D].  One 16x16 tile per wave, K=4096.
// ---------------------------------------------------------------------------
__global__ __launch_bounds__(256) void k_wh(const float* __restrict__ hs,
                                            const float* __restrict__ Wm,
                                            float* __restrict__ Wh) {
    const int lane  = threadIdx.x & 31;
    const int w     = blockIdx.x * 8 + (threadIdx.x >> 5);
    const int tRow  = w >> 3;           // N/16 = 512
    const int tCol  = w & 7;            // D/16 = 8
    const int r     = lane & 15;
    const int khalf = (lane >> 4) << 1; // 0 or 2

    const float* aRow = hs + (size_t)(tRow * 16 + r) * M;
    const float* bCol = Wm + (tCol * 16 + r);

    v8f acc = {};
#pragma unroll 4
    for (int k = 0; k < M; k += 4) {
        const int kk = k + khalf;
        v2f a = *(const v2f*)(aRow + kk);       // h_s[row, kk..kk+1], 8B aligned
        v2f b;
        b.x = bCol[(size_t)kk * D];             // W[kk,   col]
        b.y = bCol[(size_t)(kk + 1) * D];       // W[kk+1, col]
        acc = wmma_f32(a, b, acc);
    }

    const int rowBase = tRow * 16 + ((lane >> 4) << 3);
    const int col     = tCol * 16 + (lane & 15);
#pragma unroll
    for (int v = 0; v < 8; ++v)
        Wh[(size_t)(rowBase + v) * D + col] = acc[v];
}

// ---------------------------------------------------------------------------
// Kernel 2: raw logits for BOTH heads, written into att1/att2 output slots.
//   e = leaky(Wh @ h_k^T); e = mask(h_s>0) ? e : NEG
// B[k][n] = h_k[n][k]: k is the contiguous axis of h_k -> free transpose.
// ---------------------------------------------------------------------------
__global__ __launch_bounds__(256) void k_logits(const float* __restrict__ hs,
                                                const float* __restrict__ Wh,
                                                const float* __restrict__ hk1,
                                                const float* __restrict__ hk2,
                                                float* __restrict__ att1,
                                                float* __restrict__ att2) {
    const int lane  = threadIdx.x & 31;
    const int w     = blockIdx.x * 8 + (threadIdx.x >> 5);
    const int tRow  = w >> 8;           // N/16 = 512
    const int tCol  = w & 255;          // M/16 = 256
    const int r     = lane & 15;
    const int khalf = (lane >> 4) << 1;

    const float* aRow  = Wh  + (size_t)(tRow * 16 + r) * D;
    const float* b1Row = hk1 + (size_t)(tCol * 16 + r) * D;
    const float* b2Row = hk2 + (size_t)(tCol * 16 + r) * D;

    v8f acc1 = {}, acc2 = {};
#pragma unroll 4
    for (int k = 0; k < D; k += 4) {
        const int kk = k + khalf;
        v2f a  = *(const v2f*)(aRow  + kk);
        v2f b1 = *(const v2f*)(b1Row + kk);
        v2f b2 = *(const v2f*)(b2Row + kk);
        acc1 = wmma_f32(a, b1, acc1);
        acc2 = wmma_f32(a, b2, acc2);
    }

    const int rowBase = tRow * 16 + ((lane >> 4) << 3);
    const int col     = tCol * 16 + (lane & 15);
#pragma unroll
    for (int v = 0; v < 8; ++v) {
        const size_t idx = (size_t)(rowBase + v) * M + col;
        const float msk = hs[idx];
        float e1 = acc1[v]; e1 = (e1 >= 0.0f) ? e1 : LRELU_ALPHA * e1;
        float e2 = acc2[v]; e2 = (e2 >= 0.0f) ? e2 : LRELU_ALPHA * e2;
        att1[idx] = (msk > 0.0f) ? e1 : NEGV;
        att2[idx] = (msk > 0.0f) ? e2 : NEGV;
    }
}

// ---------------------------------------------------------------------------
// Kernel 3: in-place row softmax over M=4096. grid=(N, 2 heads), 256 thr/blk.
// ---------------------------------------------------------------------------
__global__ __launch_bounds__(256) void k_softmax(float* __restrict__ att1,
                                                 float* __restrict__ att2) {
    __shared__ float red[256];
    const int tid = threadIdx.x;
    float* rowp = ((blockIdx.y == 0) ? att1 : att2) + (size_t)blockIdx.x * M;

    float vals[16];
    float mx = -INFINITY;
#pragma unroll
    for (int i = 0; i < 16; ++i) {
        vals[i] = rowp[tid + i * 256];
        mx = fmaxf(mx, vals[i]);
    }
    red[tid] = mx; __syncthreads();
    for (int s = 128; s > 0; s >>= 1) {
        if (tid < s) red[tid] = fmaxf(red[tid], red[tid + s]);
        __syncthreads();
    }
    mx = red[0]; __syncthreads();

    float sum = 0.0f;
#pragma unroll
    for (int i = 0; i < 16; ++i) {
        vals[i] = __expf(vals[i] - mx);
        sum += vals[i];
    }
    red[tid] = sum; __syncthreads();
    for (int s = 128; s > 0; s >>= 1) {
        if (tid < s) red[tid] += red[tid + s];
        __syncthreads();
    }
    const float inv = 1.0f / red[0];
#pragma unroll
    for (int i = 0; i < 16; ++i)
        rowp[tid + i * 256] = vals[i] * inv;
}

// ---------------------------------------------------------------------------
// Kernel 4: out = elu( (att1*m1)@h_k1 + (att2*m2)@h_k2 ).  Two fused GEMMs,
// dropout applied on-the-fly to A fragments, ELU epilogue.
// ---------------------------------------------------------------------------
__global__ __launch_bounds__(256) void k_out(const float* __restrict__ att1,
                                             const float* __restrict__ att2,
                                             const float* __restrict__ hk1,
                                             const float* __restrict__ hk2,
                                             float* __restrict__ out) {
    const int lane  = threadIdx.x & 31;
    const int w     = blockIdx.x * 8 + (threadIdx.x >> 5);
    const int tRow  = w >> 3;           // N/16 = 512
    const int tCol  = w & 7;            // D/16 = 8
    const int r     = lane & 15;
    const int khalf = (lane >> 4) << 1;

    const int arow = tRow * 16 + r;
    const float* a1Row = att1 + (size_t)arow * M;
    const float* a2Row = att2 + (size_t)arow * M;
    const float* b1Col = hk1 + (tCol * 16 + r);
    const float* b2Col = hk2 + (tCol * 16 + r);

    v8f acc1 = {}, acc2 = {};
#pragma unroll 4
    for (int k = 0; k < M; k += 4) {
        const int kk = k + khalf;
        v2f a1 = *(const v2f*)(a1Row + kk);
        v2f a2 = *(const v2f*)(a2Row + kk);
        const unsigned base = (unsigned)(arow * M + kk);
        a1.x *= dropmul(base,     SEED1);
        a1.y *= dropmul(base + 1, SEED1);
        a2.x *= dropmul(base,     SEED2);
        a2.y *= dropmul(base + 1, SEED2);
        v2f b1, b2;
        b1.x = b1Col[(size_t)kk * D];
        b1.y = b1Col[(size_t)(kk + 1) * D];
        b2.x = b2Col[(size_t)kk * D];
        b2.y = b2Col[(size_t)(kk + 1) * D];
        acc1 = wmma_f32(a1, b1, acc1);
        acc2 = wmma_f32(a2, b2, acc2);
    }

    const int rowBase = tRow * 16 + ((lane >> 4) << 3);
    const int col     = tCol * 16 + (lane & 15);
#pragma unroll
    for (int v = 0; v < 8; ++v) {
        const float s = acc1[v] + acc2[v];
        out[(size_t)(rowBase + v) * D + col] = (s > 0.0f) ? s : expm1f(s);
    }
}

// ---------------------------------------------------------------------------
extern "C" void kernel_launch(void* const* d_in, const int* in_sizes, int n_in,
                              void* d_out, int out_size, void* d_ws, size_t ws_size,
                              hipStream_t stream) {
    const float* h_s  = (const float*)d_in[0];
    const float* h_k1 = (const float*)d_in[1];
    const float* h_k2 = (const float*)d_in[2];
    const float* W    = (const float*)d_in[3];

    float* out  = (float*)d_out;                 // [N, D]
    float* att1 = out  + (size_t)N * D;          // [N, M]
    float* att2 = att1 + (size_t)N * M;          // [N, M]
    float* Wh   = (float*)d_ws;                  // [N, D] scratch (4 MiB)

    // 1) Wh = h_s @ W            (512*8 wave-tiles, 8 waves/block)
    k_wh<<<dim3((N / 16) * (D / 16) / 8), dim3(256), 0, stream>>>(h_s, W, Wh);

    // 2) raw masked leaky logits for both heads -> att slots
    k_logits<<<dim3((N / 16) * (M / 16) / 8), dim3(256), 0, stream>>>(
        h_s, Wh, h_k1, h_k2, att1, att2);

    // 3) in-place row softmax (grid.y selects head)
    k_softmax<<<dim3(N, 2), dim3(256), 0, stream>>>(att1, att2);

    // 4) dropout-weighted attention GEMMs + ELU epilogue
    k_out<<<dim3((N / 16) * (D / 16) / 8), dim3(256), 0, stream>>>(
        att1, att2, h_k1, h_k2, out);
}